// EyeWaveLoss_70531952935544
// MI455X (gfx1250) — compile-verified
//
#include <hip/hip_runtime.h>
#include <hip/hip_bf16.h>
#include <math.h>

#define B_    16
#define FH    256
#define FW    256
#define HW    (FH*FW)
#define CIN   32
#define NT    256
#define EPSF  1e-7f
#define PI_F  3.14159265358979323846f

typedef __attribute__((ext_vector_type(2))) float v2f;
typedef __attribute__((ext_vector_type(8))) float v8f;

__device__ __forceinline__ float sigmoidf_(float x){ return 1.f/(1.f+__expf(-x)); }
// numerically stable softplus: log(1+e^x)
__device__ __forceinline__ float softplusf_(float x){ return fmaxf(x,0.f) + log1pf(__expf(-fabsf(x))); }

// ---------------- win-map init ----------------
__global__ __launch_bounds__(256) void k_init_win(int* __restrict__ win){
  int i = blockIdx.x*256 + threadIdx.x;
  if (i < B_*HW) win[i] = -1;
}

// ---------------- FCOS-style assignment: scatter-max unique priority ----------------
__global__ __launch_bounds__(256) void k_assign(const float* __restrict__ tg,
                                                int* __restrict__ win){
  int n = blockIdx.x*256 + threadIdx.x;
  if (n >= NT) return;
  const float* t = tg + n*6;
  int   bi = (int)t[0];
  float cx = t[1], cy = t[2], w = t[3], h = t[4];
  int gi = (int)(cx * (float)FW);
  int gj = (int)(cy * (float)FH);
  int base = bi * HW;
  #pragma unroll
  for (int k = 0; k < 9; ++k){
    int nx = gi + (k/3) - 1;          // dx = [-1,-1,-1,0,0,0,1,1,1]
    int ny = gj + (k%3) - 1;          // dy = [-1,0,1,...]
    if (nx < 0 || nx >= FW || ny < 0 || ny >= FH) continue;
    float ccx = ((float)nx + 0.5f) * (1.f/(float)FW);
    float ccy = ((float)ny + 0.5f) * (1.f/(float)FH);
    if (fabsf(ccx - cx) <= 0.5f*w && fabsf(ccy - cy) <= 0.5f*h)
      atomicMax(&win[base + ny*FW + nx], n*10 + k);
  }
  int ci = min(gi, FW-1), cj = min(gj, FH-1);   // unconditional centre cell
  atomicMax(&win[base + cj*FW + ci], n*10 + 9);
}

// ---------------- attention loss: WMMA channel-sum + Gaussian scatter-max MSE ----------------
// one block per (b, y) row; 256 threads = 8 waves; each wave owns 32 pixels
__global__ __launch_bounds__(256) void k_attn(const float* __restrict__ sal,
                                              const float* __restrict__ tg,
                                              float* __restrict__ part){
  __shared__ float s_tx[NT], s_ty[NT];
  __shared__ int   s_cnt;
  __shared__ float s_sum[256];
  __shared__ float s_red[256];
  int tid = threadIdx.x;
  int b   = blockIdx.x >> 8;
  int y   = blockIdx.x & 255;
  if (tid == 0) s_cnt = 0;
  __syncthreads();
  { // compact this batch's targets into LDS (order irrelevant: max is commutative)
    int bi = (int)tg[tid*6 + 0];
    if (bi == b){
      int p = atomicAdd(&s_cnt, 1);
      s_tx[p] = tg[tid*6 + 1] * (float)FW;
      s_ty[p] = tg[tid*6 + 2] * (float)FH;
    }
  }
  __syncthreads();
  int cnt = s_cnt;

  const float* srow = sal + ((size_t)b*CIN)*HW + (size_t)y*FW;

#if __has_builtin(__builtin_amdgcn_wmma_f32_16x16x4_f32)
  { // exact-f32 channel reduction on the matrix pipe: D = A(16px x 4ch) * ones(4x16) + C
    int lane  = tid & 31, wv = tid >> 5;
    int m     = lane & 15;
    int cpair = (lane >> 4) * 2;            // A layout: lanes 0-15 hold K=0,1; 16-31 hold K=2,3
    v2f ones; ones[0] = 1.f; ones[1] = 1.f;
    #pragma unroll
    for (int st = 0; st < 2; ++st){
      int px = wv*32 + st*16 + m;
      v8f acc = {};
      #pragma unroll
      for (int kk = 0; kk < 8; ++kk){       // 8 x K=4 => 32 channels
        int ch = kk*4 + cpair;
        v2f a;
        a[0] = srow[(size_t)ch      * HW + px];
        a[1] = srow[(size_t)(ch+1)  * HW + px];
        acc = __builtin_amdgcn_wmma_f32_16x16x4_f32(false, a, false, ones,
                                                    (short)0, acc, false, false);
      }
      // every column of D identical; col 0: lane 0 -> rows 0-7, lane 16 -> rows 8-15
      if (lane == 0){
        #pragma unroll
        for (int i = 0; i < 8; ++i) s_sum[wv*32 + st*16 + i] = acc[i];
      } else if (lane == 16){
        #pragma unroll
        for (int i = 0; i < 8; ++i) s_sum[wv*32 + st*16 + 8 + i] = acc[i];
      }
    }
  }
#else
  { float s = 0.f;
    #pragma unroll 8
    for (int c = 0; c < CIN; ++c) s += srow[(size_t)c*HW + tid];
    s_sum[tid] = s; }
#endif
  __syncthreads();

  float mean = s_sum[tid] * (1.f/32.f);
  float fx = (float)tid, fy = (float)y;
  float mk = 0.f;
  for (int j = 0; j < cnt; ++j){
    float dx = fx - s_tx[j], dy = fy - s_ty[j];
    mk = fmaxf(mk, __expf(-(dx*dx + dy*dy) * (1.f/4.5f)));   // 2*sigma^2 = 4.5
  }
  float d = mean - mk;
  s_red[tid] = d*d;
  __syncthreads();
  for (int off = 128; off > 0; off >>= 1){
    if (tid < off) s_red[tid] += s_red[tid + off];
    __syncthreads();
  }
  if (tid == 0) part[blockIdx.x] = s_red[0];
}

// ---------------- per-cell losses: focal conf everywhere, CIoU + NLL on positives ----------------
__global__ __launch_bounds__(256) void k_main(const float* __restrict__ rg,
                                              const float* __restrict__ tg,
                                              const int*   __restrict__ win,
                                              float* __restrict__ part4){
  int tid = threadIdx.x;
  int idx = blockIdx.x*256 + tid;                 // < B*HW
  const float* r = rg + (size_t)idx*7;
  __builtin_prefetch(r + 7*256, 0, 0);            // global_prefetch_b8: next block-row
  float r0=r[0], r1=r[1], r2=r[2], r3=r[3], z=r[4], l0=r[5], l1=r[6];
  int  wv  = win[idx];
  bool obj = wv >= 0;

  float conf = sigmoidf_(z);
  float bce  = obj ? softplusf_(-z) : softplusf_(z);
  float pt   = obj ? conf : 1.f - conf;
  float at   = obj ? 0.25f : 0.75f;
  float omp  = 1.f - pt;
  float cterm = at * omp * omp * bce;

  float ciou = 0.f, nll = 0.f, np = 0.f;
  if (obj){
    np = 1.f;
    const float* t = tg + (wv/10)*6;
    float tcx=t[1], tcy=t[2], tw=t[3], th=t[4];
    int cls = (int)t[5];
    int gx = idx & 255, gy = (idx >> 8) & 255;
    float pcx = (sigmoidf_(r0) + (float)gx) * (1.f/256.f);
    float pcy = (sigmoidf_(r1) + (float)gy) * (1.f/256.f);
    float pw  = __expf(fminf(fmaxf(r2, -4.f), 4.f)) * (1.f/256.f);
    float ph  = __expf(fminf(fmaxf(r3, -4.f), 4.f)) * (1.f/256.f);
    float px1 = pcx - 0.5f*pw, px2 = pcx + 0.5f*pw;
    float py1 = pcy - 0.5f*ph, py2 = pcy + 0.5f*ph;
    float tx1 = tcx - 0.5f*tw, tx2 = tcx + 0.5f*tw;
    float ty1 = tcy - 0.5f*th, ty2 = tcy + 0.5f*th;
    float iw = fmaxf(fminf(px2,tx2) - fmaxf(px1,tx1), 0.f);
    float ih = fmaxf(fminf(py2,ty2) - fmaxf(py1,ty1), 0.f);
    float inter = iw*ih;
    float uni   = pw*ph + tw*th - inter;
    float iou   = inter / (uni + EPSF);
    float cw  = fmaxf(px2,tx2) - fminf(px1,tx1);
    float chh = fmaxf(py2,ty2) - fminf(py1,ty1);
    float c2  = cw*cw + chh*chh + EPSF;
    float dd  = (pcx-tcx)*(pcx-tcx) + (pcy-tcy)*(pcy-tcy);
    float da  = atanf(tw/(th+EPSF)) - atanf(pw/(ph+EPSF));
    float v   = (4.f/(PI_F*PI_F)) * da * da;
    float alpha = v / (1.f - iou + v + EPSF);
    ciou = 1.f - iou + dd/c2 + alpha*v;
    float mx  = fmaxf(l0, l1);
    float lse = mx + __logf(__expf(l0-mx) + __expf(l1-mx));
    nll = lse - (cls == 0 ? l0 : l1);
  }

  __shared__ float sA[256], sB[256], sC[256], sD[256];
  sA[tid]=cterm; sB[tid]=ciou; sC[tid]=nll; sD[tid]=np;
  __syncthreads();
  for (int off = 128; off > 0; off >>= 1){
    if (tid < off){
      sA[tid]+=sA[tid+off]; sB[tid]+=sB[tid+off];
      sC[tid]+=sC[tid+off]; sD[tid]+=sD[tid+off];
    }
    __syncthreads();
  }
  if (tid == 0){
    float4 o; o.x=sA[0]; o.y=sB[0]; o.z=sC[0]; o.w=sD[0];
    ((float4*)part4)[blockIdx.x] = o;
  }
}

// ---------------- deterministic final combine ----------------
__global__ __launch_bounds__(256) void k_final(const float* __restrict__ attn_part,
                                               const float* __restrict__ part4,
                                               float* __restrict__ out){
  __shared__ float s0[256], s1[256], s2[256], s3[256], s4[256];
  int tid = threadIdx.x;
  float a=0.f, c=0.f, u=0.f, l=0.f, p=0.f;
  for (int i = tid; i < 4096; i += 256){
    a += attn_part[i];
    float4 q = ((const float4*)part4)[i];
    c += q.x; u += q.y; l += q.z; p += q.w;
  }
  s0[tid]=a; s1[tid]=c; s2[tid]=u; s3[tid]=l; s4[tid]=p;
  __syncthreads();
  for (int off = 128; off > 0; off >>= 1){
    if (tid < off){
      s0[tid]+=s0[tid+off]; s1[tid]+=s1[tid+off]; s2[tid]+=s2[tid+off];
      s3[tid]+=s3[tid+off]; s4[tid]+=s4[tid+off];
    }
    __syncthreads();
  }
  if (tid == 0){
    float norm = fmaxf(s4[0], 1.f);
    float attn_mean = s0[0] * (1.f/(float)(B_*FH*FW));
    out[0] = 2.f*(s2[0]/norm) + (s1[0]/norm) + (s3[0]/norm) + 5.f*attn_mean;
  }
}

extern "C" void kernel_launch(void* const* d_in, const int* in_sizes, int n_in,
                              void* d_out, int out_size, void* d_ws, size_t ws_size,
                              hipStream_t stream){
  const float* rg  = (const float*)d_in[0];   // raw_grid (16,65536,7) f32
  const float* sal = (const float*)d_in[1];   // sal (16,32,256,256)   f32
  const float* tg  = (const float*)d_in[2];   // targets (256,6)       f32
  char* ws = (char*)d_ws;
  int*   win       = (int*)ws;                                  // 4 MB
  float* attn_part = (float*)(ws + (size_t)B_*HW*4);            // 16 KB
  float* part4     = (float*)(ws + (size_t)B_*HW*4 + 4096*4);   // 64 KB (16B aligned)

  hipLaunchKernelGGL(k_init_win, dim3(4096), dim3(256), 0, stream, win);
  hipLaunchKernelGGL(k_assign,   dim3(1),    dim3(256), 0, stream, tg, win);
  hipLaunchKernelGGL(k_attn,     dim3(B_*FH),dim3(256), 0, stream, sal, tg, attn_part);
  hipLaunchKernelGGL(k_main,     dim3(4096), dim3(256), 0, stream, rg, tg, win, part4);
  hipLaunchKernelGGL(k_final,    dim3(1),    dim3(256), 0, stream, attn_part, part4, (float*)d_out);
}